// FuseSliceCatSameInputModule_v2_5720896438285
// MI455X (gfx1250) — compile-verified
//
#include <hip/hip_runtime.h>
#include <cstdint>

// Reference: out[g][r][j*32+k] = in[r][512*g + 64*j + k], g,j in [0,8), k in [0,32)
// in: [16384, 4096] fp32 ; d_out: 8 contiguous [16384, 256] fp32 blocks.

#define N_ROWS     16384
#define N_COLS     4096
#define GROUP_COLS 256
#define TILE_ROWS  32
#define BLOCK_THREADS 256

__global__ __launch_bounds__(BLOCK_THREADS)
void fused_slice_cat_async_kernel(const float* __restrict__ in, float* __restrict__ out) {
    // 32 rows x 256 gathered cols x 4B = 32 KB staging buffer.
    __shared__ float buf[TILE_ROWS * GROUP_COLS];

    const uint32_t g    = blockIdx.y;                 // output group 0..7
    const uint32_t row0 = blockIdx.x * TILE_ROWS;     // first row of this tile
    const uint32_t t    = threadIdx.x;                // 0..255

    // Low 32 bits of the flat shared pointer == LDS byte offset (aperture in high bits).
    const uint32_t lds_base = (uint32_t)(uintptr_t)(&buf[0]);

    const uint64_t in_base  = (uint64_t)(uintptr_t)in;
    const uint64_t out_base = (uint64_t)(uintptr_t)out
                            + (uint64_t)g * (uint64_t)(N_ROWS * GROUP_COLS) * 4ull;

    // ---- Phase 1: async gather, global -> LDS (16 B per lane per issue) ----
    // idx in [0,2048): covers 32 rows x 64 float4-quads per row.
#pragma unroll
    for (int i = 0; i < 8; ++i) {
        const uint32_t idx  = (uint32_t)i * BLOCK_THREADS + t;  // 0..2047
        const uint32_t rl   = idx >> 6;                         // local row 0..31
        const uint32_t quad = idx & 63u;                        // float4 index in gathered row
        const uint32_t j    = quad >> 3;                        // slice 0..7
        const uint32_t k4   = quad & 7u;                        // float4 within 32-col slice

        const uint64_t gaddr = in_base
            + ((uint64_t)(row0 + rl) * N_COLS + g * 512u + j * 64u + k4 * 4u) * 4ull;
        const uint32_t laddr = lds_base + idx * 16u;

        // VDST = LDS byte address (VGPR), VADDR = 64-bit global address, saddr = off
        asm volatile("global_load_async_to_lds_b128 %0, %1, off"
                     :: "v"(laddr), "v"(gaddr)
                     : "memory");
    }

    // All async loads for this wave have landed in LDS.
#if defined(__has_builtin) && __has_builtin(__builtin_amdgcn_s_wait_asynccnt)
    __builtin_amdgcn_s_wait_asynccnt(0);
#else
    asm volatile("s_wait_asynccnt 0" ::: "memory");
#endif

    // ---- Phase 2: async scatter-out, LDS -> global (contiguous per group) ----
#pragma unroll
    for (int i = 0; i < 8; ++i) {
        const uint32_t idx  = (uint32_t)i * BLOCK_THREADS + t;
        const uint32_t rl   = idx >> 6;
        const uint32_t quad = idx & 63u;

        const uint64_t oaddr = out_base
            + ((uint64_t)(row0 + rl) * GROUP_COLS + quad * 4u) * 4ull;
        const uint32_t laddr = lds_base + idx * 16u;

        // VADDR = 64-bit global address, VSRC = LDS byte address (VGPR), saddr = off
        asm volatile("global_store_async_from_lds_b128 %0, %1, off"
                     :: "v"(oaddr), "v"(laddr)
                     : "memory");
    }

    // Drain outstanding async stores before the wave retires / LDS is reused.
#if defined(__has_builtin) && __has_builtin(__builtin_amdgcn_s_wait_asynccnt)
    __builtin_amdgcn_s_wait_asynccnt(0);
#else
    asm volatile("s_wait_asynccnt 0" ::: "memory");
#endif
}

extern "C" void kernel_launch(void* const* d_in, const int* in_sizes, int n_in,
                              void* d_out, int out_size, void* d_ws, size_t ws_size,
                              hipStream_t stream) {
    (void)in_sizes; (void)n_in; (void)d_ws; (void)ws_size; (void)out_size;

    const float* in  = (const float*)d_in[0];   // [16384, 4096] fp32
    float*       out = (float*)d_out;           // 8 x [16384, 256] fp32, concatenated

    dim3 grid(N_ROWS / TILE_ROWS, 8, 1);        // (512 row tiles, 8 groups)
    dim3 block(BLOCK_THREADS, 1, 1);
    fused_slice_cat_async_kernel<<<grid, block, 0, stream>>>(in, out);
}